// Neighbor_Interaction_Aggregation_14568529068221
// MI455X (gfx1250) — compile-verified
//
#include <hip/hip_runtime.h>

#define D_FEAT 128
#define WAVES_PER_BLOCK 8
#define EDGES_PER_WAVE  8
#define EDGES_PER_BLOCK (WAVES_PER_BLOCK * EDGES_PER_WAVE)
#define THREADS 256

typedef float v4f __attribute__((ext_vector_type(4)));
typedef int   v4i __attribute__((ext_vector_type(4)));
typedef __attribute__((address_space(3))) v4i lds_v4i;

// ---------------- CDNA5 async global->LDS copy (ASYNCcnt path) ----------------
#if defined(__HIP_DEVICE_COMPILE__) && __has_builtin(__builtin_amdgcn_global_load_async_to_lds_b128)
#define ASYNC_COPY_B128(gsrc, ldst)                                         \
  __builtin_amdgcn_global_load_async_to_lds_b128(                           \
      (v4i*)(gsrc), (lds_v4i*)(ldst), 0, /*th=NT*/ 1)
#else
__device__ __forceinline__ void __async_copy_b128(const void* gsrc, void* ldst) {
  unsigned loff =
      (unsigned)(unsigned long long)(__attribute__((address_space(3))) char*)ldst;
  unsigned long long ga = (unsigned long long)gsrc;
  asm volatile("global_load_async_to_lds_b128 %0, %1, off" ::"v"(loff), "v"(ga)
               : "memory");
}
#define ASYNC_COPY_B128(gsrc, ldst) __async_copy_b128((const void*)(gsrc), (void*)(ldst))
#endif

#if defined(__HIP_DEVICE_COMPILE__) && __has_builtin(__builtin_amdgcn_s_wait_asynccnt)
#define WAIT_ASYNCCNT_0() __builtin_amdgcn_s_wait_asynccnt(0)
#define WAIT_ASYNCCNT_1() __builtin_amdgcn_s_wait_asynccnt(1)
#else
#define WAIT_ASYNCCNT_0() asm volatile("s_wait_asynccnt 0x0" ::: "memory")
#define WAIT_ASYNCCNT_1() asm volatile("s_wait_asynccnt 0x1" ::: "memory")
#endif

// Hardware f32 global atomic add, no-return form (STOREcnt), device scope so
// the read-modify-write is performed at L2 (coherent across all WGPs).
__device__ __forceinline__ void atomic_add_f32(float* p, float v) {
  asm volatile("global_atomic_add_f32 %0, %1, off scope:SCOPE_DEV"
               ::"v"((unsigned long long)(__SIZE_TYPE__)p), "v"(v)
               : "memory");
}

// ---------------------------------------------------------------------------
// e-branch: e_h[dst] += e_sim[e,1] * feat[src].  One wave per edge; each lane
// owns 4 of the 128 features (16B gather + 4 scatter atomics).
// ---------------------------------------------------------------------------
__global__ __launch_bounds__(THREADS) void edge_e_kernel(
    const float* __restrict__ e_sim, const float* __restrict__ feat,
    const int* __restrict__ src, const int* __restrict__ dst,
    float* __restrict__ e_h, int n_edges) {
  const int lane = threadIdx.x & 31;
  const int wave = threadIdx.x >> 5;
  long long e = (long long)blockIdx.x * EDGES_PER_BLOCK + wave;
  for (int t = 0; t < EDGES_PER_WAVE; ++t, e += WAVES_PER_BLOCK) {
    if (e >= n_edges) break;
    // use-once strided column read: keep it non-temporal
    const float w = __builtin_nontemporal_load(&e_sim[e * D_FEAT + 1]);
    const int s = src[e];
    const int d = dst[e];
    const v4f z =
        *reinterpret_cast<const v4f*>(&feat[(long long)s * D_FEAT + lane * 4]);
    float* o = &e_h[(long long)d * D_FEAT + lane * 4];
    atomic_add_f32(o + 0, w * z.x);
    atomic_add_f32(o + 1, w * z.y);
    atomic_add_f32(o + 2, w * z.z);
    atomic_add_f32(o + 3, w * z.w);
  }
}

// ---------------------------------------------------------------------------
// i-branch: i_h[dst] += e_cor[e,:] * feat[src] (elementwise).  The 512B e_cor
// rows are streamed into LDS with double-buffered async copies (ASYNCcnt),
// hiding the sequential stream behind the random feat gathers. Each wave owns
// its own two 512B LDS slots, so no cross-wave barriers are needed.
// ---------------------------------------------------------------------------
__global__ __launch_bounds__(THREADS) void edge_i_kernel(
    const float* __restrict__ e_cor, const float* __restrict__ feat,
    const int* __restrict__ src, const int* __restrict__ dst,
    float* __restrict__ i_h, int n_edges) {
  __shared__ __align__(16) float tile[2][WAVES_PER_BLOCK][D_FEAT];
  const int lane = threadIdx.x & 31;
  const int wave = threadIdx.x >> 5;
  long long e = (long long)blockIdx.x * EDGES_PER_BLOCK + wave;

  if (e < n_edges)  // prefetch iteration 0 (each lane moves 16B of the row)
    ASYNC_COPY_B128(&e_cor[e * D_FEAT + lane * 4], &tile[0][wave][lane * 4]);

  for (int t = 0; t < EDGES_PER_WAVE; ++t) {
    const long long en = e + WAVES_PER_BLOCK;
    const int cur = t & 1;
    const bool have_next = (t + 1 < EDGES_PER_WAVE) && (en < (long long)n_edges);
    if (have_next)  // issue next row before consuming current one
      ASYNC_COPY_B128(&e_cor[en * D_FEAT + lane * 4], &tile[cur ^ 1][wave][lane * 4]);
    // async loads complete in order: <=1 outstanding means 'cur' has landed
    if (have_next) { WAIT_ASYNCCNT_1(); } else { WAIT_ASYNCCNT_0(); }
    if (e < n_edges) {
      const int s = src[e];
      const int d = dst[e];
      const v4f z =
          *reinterpret_cast<const v4f*>(&feat[(long long)s * D_FEAT + lane * 4]);
      const v4f w = *reinterpret_cast<const v4f*>(&tile[cur][wave][lane * 4]);
      float* o = &i_h[(long long)d * D_FEAT + lane * 4];
      atomic_add_f32(o + 0, w.x * z.x);
      atomic_add_f32(o + 1, w.y * z.y);
      atomic_add_f32(o + 2, w.z * z.z);
      atomic_add_f32(o + 3, w.w * z.w);
    }
    e = en;
  }
}

// ---------------------------------------------------------------------------
// combine: out = relu(e_h * i_h) + self_alpha[:,1] * feat
// (0.5((a+b)^2 - a^2 - b^2) == a*b exactly)
// ---------------------------------------------------------------------------
__global__ __launch_bounds__(THREADS) void combine_kernel(
    const float* __restrict__ e_h, const float* __restrict__ i_h,
    const float* __restrict__ feat, const float* __restrict__ self_alpha,
    float* __restrict__ out, long long total4) {
  const long long q = (long long)blockIdx.x * THREADS + threadIdx.x;
  if (q >= total4) return;
  const long long idx = q * 4;
  const long long node = idx >> 7;  // /128
  const float a = self_alpha[node * 2 + 1];
  // accumulators are dead after this kernel: read non-temporally (don't pin L2)
  const v4f eh = __builtin_nontemporal_load(reinterpret_cast<const v4f*>(e_h + idx));
  const v4f ih = __builtin_nontemporal_load(reinterpret_cast<const v4f*>(i_h + idx));
  const v4f z  = *reinterpret_cast<const v4f*>(feat + idx);
  v4f r;
  r.x = fmaxf(eh.x * ih.x, 0.0f) + a * z.x;
  r.y = fmaxf(eh.y * ih.y, 0.0f) + a * z.y;
  r.z = fmaxf(eh.z * ih.z, 0.0f) + a * z.z;
  r.w = fmaxf(eh.w * ih.w, 0.0f) + a * z.w;
  __builtin_nontemporal_store(r, reinterpret_cast<v4f*>(out + idx));
}

extern "C" void kernel_launch(void* const* d_in, const int* in_sizes, int n_in,
                              void* d_out, int out_size, void* d_ws, size_t ws_size,
                              hipStream_t stream) {
  const float* feat       = (const float*)d_in[0];
  const float* self_alpha = (const float*)d_in[1];
  const float* e_sim      = (const float*)d_in[2];
  const float* e_cor      = (const float*)d_in[3];
  const int*   src_sim    = (const int*)d_in[4];
  const int*   dst_sim    = (const int*)d_in[5];
  const int*   src_cor    = (const int*)d_in[6];
  const int*   dst_cor    = (const int*)d_in[7];
  float* out = (float*)d_out;

  const long long n_nodes = (long long)in_sizes[0] / D_FEAT;  // 100000
  const int       n_edges = in_sizes[4];                      // 640000

  float* e_h = (float*)d_ws;
  float* i_h = e_h + (size_t)n_nodes * D_FEAT;

  // accumulators must start at zero on every call (atomic accumulation)
  (void)hipMemsetAsync(d_ws, 0, (size_t)(2 * n_nodes * D_FEAT) * sizeof(float),
                       stream);

  const int edge_blocks = (n_edges + EDGES_PER_BLOCK - 1) / EDGES_PER_BLOCK;
  edge_e_kernel<<<edge_blocks, THREADS, 0, stream>>>(e_sim, feat, src_sim, dst_sim,
                                                     e_h, n_edges);
  edge_i_kernel<<<edge_blocks, THREADS, 0, stream>>>(e_cor, feat, src_cor, dst_cor,
                                                     i_h, n_edges);

  const long long total4 = n_nodes * (D_FEAT / 4);
  const int cblocks = (int)((total4 + THREADS - 1) / THREADS);
  combine_kernel<<<cblocks, THREADS, 0, stream>>>(e_h, i_h, feat, self_alpha, out,
                                                  total4);
}